// GATLayer_2731599200750
// MI455X (gfx1250) — compile-verified
//
#include <hip/hip_runtime.h>
#include <cstdint>

#define N_NODES   100000
#define E_EDGES   1600000
#define E_TOT     (E_EDGES + N_NODES)   /* 1,700,000 edges incl. self loops */
#define IN_DIM    128
#define HC        256                    /* H*C */
#define NHEAD     4
#define NEG_SLOPE 0.2f
#define BN_EPS    1e-5f

typedef __attribute__((ext_vector_type(2))) float v2f;
typedef __attribute__((ext_vector_type(8))) float v8f;

// ---------- order-preserving float<->uint encoding for atomic max ----------
__device__ __forceinline__ unsigned enc_f32(float f) {
  unsigned u = __float_as_uint(f);
  return (u & 0x80000000u) ? ~u : (u | 0x80000000u);
}
__device__ __forceinline__ float dec_f32(unsigned u) {
  unsigned b = (u & 0x80000000u) ? (u ^ 0x80000000u) : ~u;
  return __uint_as_float(b);
}

// ---------- generic zero ----------
__global__ void zero_f32(float* __restrict__ p, long long n) {
  long long i = (long long)blockIdx.x * blockDim.x + threadIdx.x;
  if (i < n) p[i] = 0.0f;
}

// ---------- projection GEMM: out[N,256] = A[N,128] @ W[128,256] + b ----------
// One wave per 16x16 output tile, K swept by 32 chained V_WMMA_F32_16X16X4_F32.
__global__ void proj_gemm_wmma(const float* __restrict__ A,
                               const float* __restrict__ W,
                               const float* __restrict__ bvec,
                               float* __restrict__ out) {
  int wave = (int)((blockIdx.x * blockDim.x + threadIdx.x) >> 5);
  int lane = threadIdx.x & 31;
  int nTile = wave & 15;          // 256/16 = 16 N tiles
  int mTile = wave >> 4;
  if (mTile * 16 >= N_NODES) return;   // wave-uniform: EXEC stays all-ones

  int half = lane >> 4;           // 0: lanes 0-15, 1: lanes 16-31
  int mr   = lane & 15;           // row (A) / col (B) within tile

  // A frag: lane holds A[m0+mr][kb + 2*half + {0,1}]  -> contiguous pair
  const float* arow = A + (size_t)(mTile * 16 + mr) * IN_DIM + half * 2;
  // B frag: lane holds W[kb + 2*half + {0,1}][n0 + mr] -> stride HC
  const float* bcol = W + (size_t)(half * 2) * HC + nTile * 16 + mr;

  v8f acc = {};
#pragma unroll
  for (int kk = 0; kk < IN_DIM / 4; ++kk) {
    v2f a = *(const v2f*)(arow + kk * 4);          // 8B aligned global_load_b64
    v2f b;
    b.x = bcol[(size_t)(kk * 4) * HC];
    b.y = bcol[(size_t)(kk * 4) * HC + HC];
    acc = __builtin_amdgcn_wmma_f32_16x16x4_f32(
        /*neg_a=*/false, a, /*neg_b=*/false, b,
        /*c_mod=*/(short)0, acc, /*reuse_a=*/false, /*reuse_b=*/false);
  }

  float bias = bvec[nTile * 16 + mr];
  // D layout: VGPR r -> M = r + 8*half, N = mr
  float* orow = out + (size_t)(mTile * 16 + 8 * half) * HC + nTile * 16 + mr;
#pragma unroll
  for (int r = 0; r < 8; ++r) orow[(size_t)r * HC] = acc[r] + bias;
}

// ---------- per-edge attention score + segment max ----------
// One wave per edge. Lane l owns channels [8l, 8l+8) -> head h = l/8.
__global__ void edge_alpha_max(const float* __restrict__ xl,
                               const float* __restrict__ xr,
                               const long long* __restrict__ ei,
                               const float* __restrict__ att,
                               float* __restrict__ alpha,
                               unsigned* __restrict__ amax_enc) {
  long long wave = ((long long)blockIdx.x * blockDim.x + threadIdx.x) >> 5;
  if (wave >= (long long)E_TOT) return;
  int lane = threadIdx.x & 31;

  long long src, dst;
  if (wave < (long long)E_EDGES) { src = ei[wave]; dst = ei[(long long)E_EDGES + wave]; }
  else                           { src = wave - E_EDGES; dst = src; }

  int cbase = lane * 8;
  const float4* pl = (const float4*)(xl + src * HC + cbase);
  const float4* pr = (const float4*)(xr + dst * HC + cbase);
  const float4* pa = (const float4*)(att + cbase);

  float s = 0.0f;
#pragma unroll
  for (int q = 0; q < 2; ++q) {
    float4 l4 = pl[q], r4 = pr[q], a4 = pa[q];
    float v;
    v = l4.x + r4.x; s += a4.x * (v > 0.0f ? v : NEG_SLOPE * v);
    v = l4.y + r4.y; s += a4.y * (v > 0.0f ? v : NEG_SLOPE * v);
    v = l4.z + r4.z; s += a4.z * (v > 0.0f ? v : NEG_SLOPE * v);
    v = l4.w + r4.w; s += a4.w * (v > 0.0f ? v : NEG_SLOPE * v);
  }
  // reduce within each aligned group of 8 lanes (one head)
  s += __shfl_xor(s, 1);
  s += __shfl_xor(s, 2);
  s += __shfl_xor(s, 4);

  if ((lane & 7) == 0) {
    int h = lane >> 3;
    alpha[wave * NHEAD + h] = s;
    atomicMax(&amax_enc[dst * NHEAD + h], enc_f32(s));   // global_atomic_max_u32
  }
}

// ---------- exp(alpha - amax) + segment sum ----------
__global__ void edge_exp_denom(float* __restrict__ alpha,
                               const long long* __restrict__ ei,
                               const unsigned* __restrict__ amax_enc,
                               float* __restrict__ denom) {
  long long idx = (long long)blockIdx.x * blockDim.x + threadIdx.x;
  if (idx >= (long long)E_TOT * NHEAD) return;
  long long e = idx >> 2;
  int h = (int)(idx & 3);
  long long dst = (e < (long long)E_EDGES) ? ei[(long long)E_EDGES + e] : (e - E_EDGES);
  float amax = dec_f32(amax_enc[dst * NHEAD + h]);
  float v = __expf(alpha[idx] - amax);
  alpha[idx] = v;
  atomicAdd(&denom[dst * NHEAD + h], v);                 // global_atomic_add_f32
}

// ---------- weighted scatter-aggregate (division by denom deferred) ----------
__global__ void edge_aggregate(const float* __restrict__ xl,
                               const float* __restrict__ alpha,
                               const long long* __restrict__ ei,
                               float* __restrict__ accum) {
  long long wave = ((long long)blockIdx.x * blockDim.x + threadIdx.x) >> 5;
  if (wave >= (long long)E_TOT) return;
  int lane = threadIdx.x & 31;

  long long src, dst;
  if (wave < (long long)E_EDGES) { src = ei[wave]; dst = ei[(long long)E_EDGES + wave]; }
  else                           { src = wave - E_EDGES; dst = src; }

  float w = alpha[wave * NHEAD + (lane >> 3)];
  const float4* ps = (const float4*)(xl + src * HC + lane * 8);
  float* pd = accum + dst * HC + lane * 8;
  float4 v0 = ps[0], v1 = ps[1];
  atomicAdd(pd + 0, v0.x * w);
  atomicAdd(pd + 1, v0.y * w);
  atomicAdd(pd + 2, v0.z * w);
  atomicAdd(pd + 3, v0.w * w);
  atomicAdd(pd + 4, v1.x * w);
  atomicAdd(pd + 5, v1.y * w);
  atomicAdd(pd + 6, v1.z * w);
  atomicAdd(pd + 7, v1.w * w);
}

// ---------- BN statistics: per-channel sum and sum-of-squares ----------
__global__ void bn_stats(const float* __restrict__ accum,
                         const float* __restrict__ denom,
                         const float* __restrict__ bias,
                         float* __restrict__ stats) {
  int j = threadIdx.x;              // channel 0..255
  float bj = bias[j];
  float s1 = 0.0f, s2 = 0.0f;
  for (int n = blockIdx.x; n < N_NODES; n += gridDim.x) {
    float d = denom[(size_t)n * NHEAD + (j >> 6)];
    float v = accum[(size_t)n * HC + j] / (d + 1e-16f) + bj;
    s1 += v; s2 += v * v;
  }
  atomicAdd(&stats[j], s1);
  atomicAdd(&stats[HC + j], s2);
}

// ---------- BN apply + ReLU (in place on accumulation buffer) ----------
__global__ void bn_apply(float* __restrict__ out,
                         const float* __restrict__ denom,
                         const float* __restrict__ bias,
                         const float* __restrict__ stats,
                         const float* __restrict__ gamma,
                         const float* __restrict__ beta) {
  long long idx = (long long)blockIdx.x * blockDim.x + threadIdx.x;
  if (idx >= (long long)N_NODES * HC) return;
  int j = (int)(idx & (HC - 1));
  long long n = idx >> 8;
  float d = denom[n * NHEAD + (j >> 6)];
  float v = out[idx] / (d + 1e-16f) + bias[j];
  const float invN = 1.0f / (float)N_NODES;
  float mean = stats[j] * invN;
  float var  = stats[HC + j] * invN - mean * mean;
  float y = (v - mean) * rsqrtf(var + BN_EPS) * gamma[j] + beta[j];
  out[idx] = y > 0.0f ? y : 0.0f;
}

extern "C" void kernel_launch(void* const* d_in, const int* in_sizes, int n_in,
                              void* d_out, int out_size, void* d_ws, size_t ws_size,
                              hipStream_t stream) {
  const float*     feature = (const float*)d_in[0];
  const long long* eidx    = (const long long*)d_in[1];   // int64 [2, E]
  const float*     W_l     = (const float*)d_in[2];
  const float*     b_l     = (const float*)d_in[3];
  const float*     W_r     = (const float*)d_in[4];
  const float*     b_r     = (const float*)d_in[5];
  const float*     att     = (const float*)d_in[6];
  const float*     bias    = (const float*)d_in[7];
  const float*     gamma   = (const float*)d_in[8];
  const float*     beta    = (const float*)d_in[9];
  float* out = (float*)d_out;

  // workspace layout (bytes)
  char* ws = (char*)d_ws;
  float*    xl       = (float*)(ws);                                   // N*256 f32
  float*    xr       = (float*)(ws + 102400000ll);                     // N*256 f32
  float*    alpha    = (float*)(ws + 204800000ll);                     // Etot*4 f32
  unsigned* amax_enc = (unsigned*)(ws + 232000000ll);                  // N*4 u32
  float*    denom    = (float*)(ws + 233600000ll);                     // N*4 f32
  float*    stats    = (float*)(ws + 235200000ll);                     // 512 f32
  (void)ws_size; (void)in_sizes; (void)n_in; (void)out_size;

  // 1) zero accumulators: amax/denom/stats are contiguous; d_out is the aggregation buffer
  {
    long long nz = (long long)N_NODES * NHEAD * 2 + 512;               // amax + denom + stats
    zero_f32<<<(int)((nz + 255) / 256), 256, 0, stream>>>((float*)amax_enc, nz);
    long long no = (long long)N_NODES * HC;
    zero_f32<<<(int)((no + 255) / 256), 256, 0, stream>>>(out, no);
  }

  // 2) fp32 WMMA projections: x_l, x_r
  {
    int waves = (N_NODES / 16) * (HC / 16);          // 100000 waves
    int blocks = waves / 8;                          // 8 waves (256 threads) per block
    proj_gemm_wmma<<<blocks, 256, 0, stream>>>(feature, W_l, b_l, xl);
    proj_gemm_wmma<<<blocks, 256, 0, stream>>>(feature, W_r, b_r, xr);
  }

  // 3) edge scores + segment max (one wave per edge)
  {
    int blocks = (int)(((long long)E_TOT + 7) / 8);
    edge_alpha_max<<<blocks, 256, 0, stream>>>(xl, xr, eidx, att, alpha, amax_enc);
  }

  // 4) exp + denominator
  {
    long long nt = (long long)E_TOT * NHEAD;
    edge_exp_denom<<<(int)((nt + 255) / 256), 256, 0, stream>>>(alpha, eidx, amax_enc, denom);
  }

  // 5) weighted scatter-aggregate into d_out
  {
    int blocks = (int)(((long long)E_TOT + 7) / 8);
    edge_aggregate<<<blocks, 256, 0, stream>>>(xl, alpha, eidx, out);
  }

  // 6) BN batch statistics
  bn_stats<<<512, HC, 0, stream>>>(out, denom, bias, stats);

  // 7) BN apply + ReLU in place
  {
    long long nt = (long long)N_NODES * HC;
    bn_apply<<<(int)((nt + 255) / 256), 256, 0, stream>>>(out, denom, bias, stats, gamma, beta);
  }
}